// LivingTokenFoam_44624710205612
// MI455X (gfx1250) — compile-verified
//
#include <hip/hip_runtime.h>
#include <hip/hip_bf16.h>
#include <math.h>

// Problem constants (match reference)
constexpr int VV = 128000;
constexpr int DD = 128;
constexpr int NN = 4;
constexpr int KK = 4;
constexpr int TT = 16;
constexpr int SS = 5;
constexpr int NB = 256;          // partial-reduction blocks for softmax
constexpr float EPSF = 1e-10f;

typedef float v2f __attribute__((ext_vector_type(2)));
typedef float v8f __attribute__((ext_vector_type(8)));

// ---------------------------------------------------------------- reductions
__device__ inline float blockReduceSum128(float v, float* red) {
    int t = threadIdx.x;
    red[t] = v; __syncthreads();
    for (int off = 64; off > 0; off >>= 1) {
        if (t < off) red[t] += red[t + off];
        __syncthreads();
    }
    float r = red[0]; __syncthreads();
    return r;
}

// ---------------------------------------------------------------- init
__global__ void zero_kernel(float* p, int n) {
    int i = blockIdx.x * blockDim.x + threadIdx.x;
    if (i < n) p[i] = 0.0f;
}

// ---------------------------------------------------------------- foam step
// One block per foam k, 128 threads (thread == dim index d).
// Updates memory[k] in place, writes scaled state rows into A so that
// A^T A == rho_mix exactly.
__global__ __launch_bounds__(128) void foam_kernel(
    const float* __restrict__ E, const float* __restrict__ bases,
    const float* __restrict__ tension, const float* __restrict__ temperature,
    const float* __restrict__ target_sim, const float* __restrict__ step_size,
    const float* __restrict__ gate_logit, const float* __restrict__ decay_base,
    const float* __restrict__ sensitivity, const float* __restrict__ noise,
    const int* __restrict__ tokens, float* __restrict__ mem,
    float* __restrict__ A, int t)
{
    const int k = blockIdx.x;
    const int tid = threadIdx.x;

    __shared__ float st[NN][DD];
    __shared__ float ex[NN][DD];
    __shared__ float fsh[NN][DD];
    __shared__ float red[128];

    // interaction = softmax(-tension/temp) * (1 - I)
    float temp = fmaxf(fabsf(temperature[k]), 0.01f);
    float inter[NN][NN];
    for (int n = 0; n < NN; n++) {
        float row[NN]; float mx = -3.0e38f;
        for (int m = 0; m < NN; m++) {
            row[m] = -tension[(k * NN + n) * NN + m] / temp;
            mx = fmaxf(mx, row[m]);
        }
        float ssum = 0.f;
        for (int m = 0; m < NN; m++) { row[m] = expf(row[m] - mx); ssum += row[m]; }
        for (int m = 0; m < NN; m++) inter[n][m] = (n == m) ? 0.f : row[m] / ssum;
    }
    const float tgt  = target_sim[k];
    const float stp  = fminf(fmaxf(fabsf(step_size[k]), 0.001f), 0.5f);
    const float gate = 1.f / (1.f + expf(-gate_logit[k]));
    const float sens = fabsf(sensitivity[0]);
    const float dbase = decay_base[0];

    const int tok = tokens[t];
    const float x = E[(size_t)tok * DD + tid];

    float mm = 0.f;
    for (int n = 0; n < NN; n++) mm += mem[(k * NN + n) * DD + tid];
    mm *= (1.0f / NN);

    float dxm = blockReduceSum128(x * mm, red);
    float nx  = sqrtf(blockReduceSum128(x * x, red)) + EPSF;
    float nmm = sqrtf(blockReduceSum128(mm * mm, red)) + EPSF;
    float novelty = (nmm > 1e-8f) ? (1.f - dxm / (nx * nmm)) : 1.f;
    float decay = 1.f / (1.f + expf(-(dbase - sens * novelty)));

    float xm = x + decay * mm;
    for (int n = 0; n < NN; n++) st[n][tid] = xm;
    __syncthreads();

    for (int s = 0; s < SS; s++) {
        // expr[n][j] = sum_d st[n][d] * B[k,n,j,d]   (thread == j)
        for (int n = 0; n < NN; n++) {
            const float* Bn = bases + ((size_t)(k * NN + n)) * DD * DD + (size_t)tid * DD;
            float acc = 0.f;
            for (int d = 0; d < DD; d++) acc = fmaf(st[n][d], Bn[d], acc);
            ex[n][tid] = acc;
        }
        __syncthreads();

        // 4x4 Gram of expr -> cos-sims, diff norms
        float gram[NN][NN];
        for (int n = 0; n < NN; n++)
            for (int m = n; m < NN; m++) {
                float v = blockReduceSum128(ex[n][tid] * ex[m][tid], red);
                gram[n][m] = v; gram[m][n] = v;
            }
        float nrm[NN];
        for (int n = 0; n < NN; n++) nrm[n] = sqrtf(fmaxf(gram[n][n], 0.f));
        float fmag[NN][NN], invdn[NN][NN];
        for (int n = 0; n < NN; n++)
            for (int m = 0; m < NN; m++) {
                float cs = gram[n][m] / ((nrm[n] + EPSF) * (nrm[m] + EPSF));
                fmag[n][m] = (cs - tgt) * inter[n][m];
                float dn2 = fmaxf(gram[n][n] + gram[m][m] - 2.f * gram[n][m], 0.f);
                invdn[n][m] = 1.f / (sqrtf(dn2) + EPSF);
            }

        // f_shared[n][j]
        for (int n = 0; n < NN; n++) {
            float acc = 0.f;
            for (int m = 0; m < NN; m++)
                acc += fmag[n][m] * (ex[n][tid] - ex[m][tid]) * invdn[n][m];
            fsh[n][tid] = acc;
        }
        __syncthreads();

        // scale = mean ||state_n|| + EPS
        float snrm = 0.f;
        for (int n = 0; n < NN; n++)
            snrm += sqrtf(blockReduceSum128(st[n][tid] * st[n][tid], red));
        float scale = snrm * 0.25f + EPSF;

        // f_local[n][d] = sum_j fsh[n][j] * B[k,n,j,d]   (thread == d, coalesced)
        for (int n = 0; n < NN; n++) {
            const float* Bn = bases + ((size_t)(k * NN + n)) * DD * DD;
            float fl = 0.f;
            for (int j = 0; j < DD; j++) fl = fmaf(fsh[n][j], Bn[(size_t)j * DD + tid], fl);
            float epsn = noise[((size_t)((t * KK + k) * SS + s) * NN + n) * DD + tid];
            st[n][tid] += stp * fl + gate * epsn * scale * 0.01f;
        }
        __syncthreads();
    }

    // trace-normalized rows into A: rho_mix = A^T A
    float tsum = 0.f;
    for (int n = 0; n < NN; n++) tsum += st[n][tid] * st[n][tid];
    float tr = blockReduceSum128(tsum, red) + EPSF;
    float inv = 1.f / sqrtf((float)KK * tr);

    for (int n = 0; n < NN; n++) {
        A[(k * NN + n) * DD + tid] = st[n][tid] * inv;
        int mi = (k * NN + n) * DD + tid;
        mem[mi] = decay * mem[mi] + (1.f - decay) * st[n][tid];
    }
}

// ---------------------------------------------------------------- eig (rank-16)
// Nonzero eigenvalues of rho_mix = A^T A equal eigenvalues of 16x16 M = A A^T.
__global__ __launch_bounds__(256) void eig_kernel(const float* __restrict__ A,
                                                  float* __restrict__ outS)
{
    __shared__ float M[16][16];
    int tid = threadIdx.x;
    int i = tid >> 4, j = tid & 15;
    float dot = 0.f;
    for (int d = 0; d < DD; d++) dot = fmaf(A[i * DD + d], A[j * DD + d], dot);
    M[i][j] = dot;
    __syncthreads();

    if (tid == 0) {
        float m[16][16];
        for (int a = 0; a < 16; a++)
            for (int b = 0; b < 16; b++) m[a][b] = M[a][b];
        // cyclic Jacobi, eigenvalues only
        for (int sweep = 0; sweep < 12; sweep++) {
            for (int p = 0; p < 15; p++)
                for (int q = p + 1; q < 16; q++) {
                    float apq = m[p][q];
                    if (fabsf(apq) < 1e-14f) continue;
                    float tau = (m[q][q] - m[p][p]) / (2.f * apq);
                    float tt = ((tau >= 0.f) ? 1.f : -1.f) /
                               (fabsf(tau) + sqrtf(1.f + tau * tau));
                    float c = 1.f / sqrtf(1.f + tt * tt);
                    float s = tt * c;
                    for (int r = 0; r < 16; r++) {
                        float mrp = m[r][p], mrq = m[r][q];
                        m[r][p] = c * mrp - s * mrq;
                        m[r][q] = s * mrp + c * mrq;
                    }
                    for (int r = 0; r < 16; r++) {
                        float mpr = m[p][r], mqr = m[q][r];
                        m[p][r] = c * mpr - s * mqr;
                        m[q][r] = s * mpr + c * mqr;
                    }
                }
        }
        // entropy: 16 real eigenvalues + 112 clip-floor (1e-12) eigenvalues
        float lam[16]; float sum = 112.f * 1e-12f;
        for (int a = 0; a < 16; a++) { lam[a] = fmaxf(m[a][a], 1e-12f); sum += lam[a]; }
        float Sr = 0.f;
        for (int a = 0; a < 16; a++) {
            float p = lam[a] / sum;
            Sr -= p * fmaxf(logf(p), -100.f);
        }
        float q0 = 1e-12f / sum;
        Sr -= 112.f * q0 * fmaxf(logf(q0), -100.f);
        *outS = Sr;
    }
}

// ---------------------------------------------------------------- logits (WMMA)
// logits[v] = || A e_v ||^2 via P = E @ A^T (GEMM [V,128]x[128,16]) then
// row-wise sum of squares. One 16x16 fp32 tile per wave, V_WMMA_F32_16X16X4_F32.
__global__ __launch_bounds__(256) void logits_kernel(const float* __restrict__ E,
                                                     const float* __restrict__ A,
                                                     float* __restrict__ logits)
{
    __shared__ float As[16 * DD];
    for (int i = threadIdx.x; i < 16 * DD; i += 256) As[i] = A[i];
    __syncthreads();

    const int wave = threadIdx.x >> 5;
    const int lane = threadIdx.x & 31;
    const int mrow = lane & 15;      // M (A-frag) / N (B-frag) index
    const int hi   = lane >> 4;      // K-half selector
    const long rowBase = ((long)blockIdx.x * 8 + wave) * 16;
    const float* Erow = E + (rowBase + mrow) * DD;

    v8f acc = {0.f, 0.f, 0.f, 0.f, 0.f, 0.f, 0.f, 0.f};
    for (int kk = 0; kk < DD; kk += 4) {
        int k0 = kk + hi * 2;
        // A-frag 16x4 fp32: lane holds E[row][k0..k0+1]
        v2f a = *reinterpret_cast<const v2f*>(Erow + k0);
        // B-frag 4x16 fp32 (= A^T tile): lane holds A[col][k0..k0+1]
        v2f b = *reinterpret_cast<const v2f*>(&As[mrow * DD + k0]);
        acc = __builtin_amdgcn_wmma_f32_16x16x4_f32(
            false, a, false, b, (short)0, acc, false, false);
    }

    // row sum of squares: reduce across the 16 lanes holding N=0..15
    for (int r = 0; r < 8; r++) {
        float v = acc[r];
        float sq = v * v;
        for (int off = 1; off < 16; off <<= 1) sq += __shfl_xor(sq, off, 16);
        if (mrow == 0) logits[rowBase + r + hi * 8] = sq;
    }
}

// ---------------------------------------------------------------- softmax
__global__ __launch_bounds__(256) void smax_part(const float* __restrict__ logits,
                                                 float* __restrict__ pm,
                                                 float* __restrict__ ps)
{
    float mx = -3.0e38f, s = 0.f;
    for (int v = blockIdx.x * 256 + threadIdx.x; v < VV; v += 256 * NB) {
        float xv = logits[v];
        if (xv > mx) { s = s * expf(mx - xv) + 1.f; mx = xv; }
        else           s += expf(xv - mx);
    }
    __shared__ float rm[256], rs[256];
    int t = threadIdx.x;
    rm[t] = mx; rs[t] = s; __syncthreads();
    for (int off = 128; off > 0; off >>= 1) {
        if (t < off) {
            float m1 = rm[t], s1 = rs[t], m2 = rm[t + off], s2 = rs[t + off];
            float nm = fmaxf(m1, m2);
            rm[t] = nm;
            rs[t] = s1 * expf(m1 - nm) + s2 * expf(m2 - nm);
        }
        __syncthreads();
    }
    if (t == 0) { pm[blockIdx.x] = rm[0]; ps[blockIdx.x] = rs[0]; }
}

__global__ __launch_bounds__(256) void smax_comb(const float* __restrict__ pm,
                                                 const float* __restrict__ ps,
                                                 float* __restrict__ lse)
{
    __shared__ float rm[256], rs[256];
    int t = threadIdx.x;
    rm[t] = pm[t]; rs[t] = ps[t]; __syncthreads();
    for (int off = 128; off > 0; off >>= 1) {
        if (t < off) {
            float m1 = rm[t], s1 = rs[t], m2 = rm[t + off], s2 = rs[t + off];
            float nm = fmaxf(m1, m2);
            rm[t] = nm;
            rs[t] = s1 * expf(m1 - nm) + s2 * expf(m2 - nm);
        }
        __syncthreads();
    }
    if (t == 0) lse[0] = rm[0] + logf(rs[0]);
}

// probs in place over logits (both live in d_out slice); partial entropy sums
__global__ __launch_bounds__(256) void probs_kernel(float* __restrict__ pl,
                                                    const float* __restrict__ lse,
                                                    float* __restrict__ ph)
{
    const float shift = lse[0];
    float h = 0.f;
    for (int v = blockIdx.x * 256 + threadIdx.x; v < VV; v += 256 * NB) {
        float lp = pl[v] - shift;
        float p = expf(lp);
        pl[v] = p;
        h -= p * fmaxf(lp, -100.f);
    }
    __shared__ float r[256];
    int t = threadIdx.x;
    r[t] = h; __syncthreads();
    for (int off = 128; off > 0; off >>= 1) {
        if (t < off) r[t] += r[t + off];
        __syncthreads();
    }
    if (t == 0) ph[blockIdx.x] = r[0];
}

__global__ __launch_bounds__(256) void final_kernel(const float* __restrict__ ph,
                                                    float* __restrict__ tail, int t)
{
    __shared__ float r[256];
    int tid = threadIdx.x;
    r[tid] = ph[tid]; __syncthreads();
    for (int off = 128; off > 0; off >>= 1) {
        if (tid < off) r[tid] += r[tid + off];
        __syncthreads();
    }
    if (tid == 0) {
        float H = r[0];
        tail[TT + t]     = H;              // H_tok
        tail[2 * TT + t] = H - tail[t];    // F_tok = H - S_rho
    }
}

// ---------------------------------------------------------------- launch
extern "C" void kernel_launch(void* const* d_in, const int* in_sizes, int n_in,
                              void* d_out, int out_size, void* d_ws, size_t ws_size,
                              hipStream_t stream)
{
    const float* E           = (const float*)d_in[0];
    const float* bases       = (const float*)d_in[1];
    const float* tension     = (const float*)d_in[2];
    const float* temperature = (const float*)d_in[3];
    const float* target_sim  = (const float*)d_in[4];
    const float* step_size   = (const float*)d_in[5];
    const float* gate_logit  = (const float*)d_in[6];
    const float* decay_base  = (const float*)d_in[7];
    const float* sensitivity = (const float*)d_in[8];
    const float* noise       = (const float*)d_in[9];
    const int*   tokens      = (const int*)d_in[10];

    float* out = (float*)d_out;
    float* ws  = (float*)d_ws;

    float* mem  = ws;                 // K*N*D   = 2048
    float* Abuf = ws + 2048;          // 16*D    = 2048
    float* pm   = ws + 4096;          // NB
    float* ps   = pm + NB;            // NB
    float* ph   = ps + NB;            // NB
    float* lse  = ph + NB;            // 1

    zero_kernel<<<(KK * NN * DD + 255) / 256, 256, 0, stream>>>(mem, KK * NN * DD);

    float* tail = out + (size_t)TT * VV;

    for (int t = 0; t < TT; t++) {
        float* probs_t = out + (size_t)t * VV;   // logits staged here, then probs
        foam_kernel<<<KK, 128, 0, stream>>>(E, bases, tension, temperature,
                                            target_sim, step_size, gate_logit,
                                            decay_base, sensitivity, noise,
                                            tokens, mem, Abuf, t);
        eig_kernel<<<1, 256, 0, stream>>>(Abuf, tail + t);
        logits_kernel<<<VV / 128, 256, 0, stream>>>(E, Abuf, probs_t);
        smax_part<<<NB, 256, 0, stream>>>(probs_t, pm, ps);
        smax_comb<<<1, 256, 0, stream>>>(pm, ps, lse);
        probs_kernel<<<NB, 256, 0, stream>>>(probs_t, lse, ph);
        final_kernel<<<1, 256, 0, stream>>>(ph, tail, t);
    }
}